// NoiseBestKRouter_73753178407349
// MI455X (gfx1250) — compile-verified
//
#include <hip/hip_runtime.h>
#include <hip/hip_bf16.h>
#include <math.h>

// ---------------------------------------------------------------------------
// Noisy top-2 MoE router, fused.  Memory-bound: x (134 MB) read exactly once.
//   logits||noise_logits = x[16 tok] @ [w_logits;w_noise]^T   via
//   V_WMMA_F32_16X16X4_F32 accumulator chain over K=2048.
// B (packed weights) served from LDS with +4 float row padding (bank-conflict
// free b64 reads); A streamed from global as per-lane b64 loads.
// ---------------------------------------------------------------------------

typedef __attribute__((ext_vector_type(2))) float v2f;
typedef __attribute__((ext_vector_type(8))) float v8f;

#define EMB_DIM     2048
#define NUM_EXPERTS 8
#define WAVES_PER_BLOCK 8
#define TOKENS_PER_WAVE 16
#define LDS_W_STRIDE 2052            // 2048 + 4 pad floats -> conflict-free
#define LDS_D_STRIDE 17              // 16 + 1 pad for the D spill

__global__ __launch_bounds__(256) void
NoiseBestKRouter_73753178407349_kernel(const float* __restrict__ x,
                                       const float* __restrict__ noise,
                                       const float* __restrict__ w_logits,
                                       const float* __restrict__ b_logits,
                                       const float* __restrict__ w_noise,
                                       const float* __restrict__ b_noise,
                                       float* __restrict__ out_router,
                                       int* __restrict__ out_idx) {
    extern __shared__ float smem[];   // [16][2052] weights + per-wave D spill

    const int tid  = threadIdx.x;
    const int lane = tid & 31;        // wave32
    const int wave = tid >> 5;
    const int m    = lane & 15;       // row-within-tile (A) / column N (B)
    const int half = lane >> 4;       // K sub-offset selector (0 or 1)

    // ---- Stage packed weights [16][2048] -> LDS (coalesced float4) --------
    // rows 0..7 = w_logits experts, rows 8..15 = w_noise experts
    for (int i = tid; i < 16 * (EMB_DIM / 4); i += 256) {
        const int n  = i >> 9;                  // row (EMB_DIM/4 == 512)
        const int c4 = i & 511;                 // float4 column
        const float* src = (n < NUM_EXPERTS)
                               ? (w_logits + (size_t)n * EMB_DIM)
                               : (w_noise + (size_t)(n - NUM_EXPERTS) * EMB_DIM);
        const float4 v = *(const float4*)(src + c4 * 4);
        *(float4*)(&smem[n * LDS_W_STRIDE + c4 * 4]) = v;
    }
    __syncthreads();

    // ---- WMMA main loop: D[16 tok][16 col] = sum_k A(16x4) * B(4x16) ------
    const int tile = blockIdx.x * WAVES_PER_BLOCK + wave;
    const int t0   = tile * TOKENS_PER_WAVE;

    // A fragment: lane<16 -> (K=k,k+1), lane>=16 -> (K=k+2,k+3), row t0+m
    const float* xrow = x + (size_t)(t0 + m) * EMB_DIM + half * 2;
    // B fragment: same per-lane pattern, row m of packed weights in LDS
    const float* brow = &smem[m * LDS_W_STRIDE + half * 2];

    v8f c = {};
    for (int kb = 0; kb < EMB_DIM; kb += 256) {
        __builtin_prefetch(xrow + kb + 256, 0, 3);   // global_prefetch_b8, 1KB ahead
#pragma unroll 8
        for (int k = kb; k < kb + 256; k += 4) {
            v2f a = *(const v2f*)(xrow + k);          // global_load_b64
            v2f b = *(const v2f*)(brow + k);          // ds_load_b64 (bank-free)
            c = __builtin_amdgcn_wmma_f32_16x16x4_f32(
                    /*neg_a=*/false, a, /*neg_b=*/false, b,
                    /*c_mod=*/(short)0, c, /*reuse_a=*/false, /*reuse_b=*/false);
        }
    }

    // ---- Spill D to LDS so each lane can own one full token row -----------
    // c[r]: lanes 0-15 -> D[M=r][N=lane], lanes 16-31 -> D[M=r+8][N=lane-16]
    float* dsc = &smem[16 * LDS_W_STRIDE + wave * (16 * LDS_D_STRIDE)];
#pragma unroll
    for (int r = 0; r < 8; ++r)
        dsc[(r + half * 8) * LDS_D_STRIDE + m] = c[r];
    __syncthreads();

    // ---- Fused epilogue: bias + softplus-noise + top-2 + softmax + scatter
    if (lane < 16) {
        const int tok = t0 + lane;
        const float* dr = dsc + lane * LDS_D_STRIDE;     // 16 values, row `lane`
        const float* nz = noise + (size_t)tok * NUM_EXPERTS;

        float best = -INFINITY, sec = -INFINITY;
        int bi = 0, si = 0;
#pragma unroll
        for (int e = 0; e < NUM_EXPERTS; ++e) {
            const float lg = dr[e] + b_logits[e];
            const float nl = dr[NUM_EXPERTS + e] + b_noise[e];
            // numerically stable softplus
            const float sp = fmaxf(nl, 0.0f) + log1pf(expf(-fabsf(nl)));
            const float v  = fmaf(nz[e], sp, lg);
            if (v > best) { sec = best; si = bi; best = v; bi = e; }
            else if (v > sec) { sec = v; si = e; }
        }
        // softmax over the two kept values (best first, as lax.top_k returns)
        const float p0 = 1.0f / (1.0f + expf(sec - best));
        const float p1 = 1.0f - p0;

        float o[NUM_EXPERTS];
#pragma unroll
        for (int e = 0; e < NUM_EXPERTS; ++e)
            o[e] = (e == bi) ? p0 : ((e == si) ? p1 : 0.0f);

        float4* outv = (float4*)(out_router + (size_t)tok * NUM_EXPERTS);
        outv[0] = make_float4(o[0], o[1], o[2], o[3]);
        outv[1] = make_float4(o[4], o[5], o[6], o[7]);
        out_idx[tok * 2 + 0] = bi;
        out_idx[tok * 2 + 1] = si;
    }
}

extern "C" void kernel_launch(void* const* d_in, const int* in_sizes, int n_in,
                              void* d_out, int out_size, void* d_ws, size_t ws_size,
                              hipStream_t stream) {
    (void)n_in; (void)out_size; (void)d_ws; (void)ws_size;
    const float* x        = (const float*)d_in[0];
    const float* noise    = (const float*)d_in[1];
    const float* w_logits = (const float*)d_in[2];
    const float* b_logits = (const float*)d_in[3];
    const float* w_noise  = (const float*)d_in[4];
    const float* b_noise  = (const float*)d_in[5];

    const int n_tokens = in_sizes[0] / EMB_DIM;              // 16384
    float* out_router  = (float*)d_out;                      // [N, 8] f32
    int*   out_idx     = (int*)((float*)d_out + (size_t)n_tokens * NUM_EXPERTS);

    const int tokens_per_block = WAVES_PER_BLOCK * TOKENS_PER_WAVE;  // 128
    const int blocks = n_tokens / tokens_per_block;                  // 128

    const size_t shmem =
        (size_t)(16 * LDS_W_STRIDE + WAVES_PER_BLOCK * 16 * LDS_D_STRIDE) *
        sizeof(float);                                               // ~137 KB

    (void)hipFuncSetAttribute(
        (const void*)NoiseBestKRouter_73753178407349_kernel,
        hipFuncAttributeMaxDynamicSharedMemorySize, (int)shmem);

    NoiseBestKRouter_73753178407349_kernel<<<blocks, 256, shmem, stream>>>(
        x, noise, w_logits, b_logits, w_noise, b_noise, out_router, out_idx);
}